// LocalAwareEncoder_4355096839064
// MI455X (gfx1250) — compile-verified
//
#include <hip/hip_runtime.h>

#define D 128

typedef __attribute__((ext_vector_type(2))) float v2f;
typedef __attribute__((ext_vector_type(8))) float v8f;

// ---------------------------------------------------------------- zero fill
__global__ __launch_bounds__(256) void zero_f4(float4* __restrict__ p, long n4) {
  long i = (long)blockIdx.x * blockDim.x + threadIdx.x;
  if (i < n4) p[i] = make_float4(0.f, 0.f, 0.f, 0.f);
}

// ------------------------------------------------- scatter: dst[didx] += v*src[sidx]
// One wave32 per edge; each lane moves a float4 (4 lanes*32 = 128 floats/row)
// and issues 4 global_atomic_add_f32 that resolve in the 192MB L2.
__global__ __launch_bounds__(256) void spmm_scatter(
    const float* __restrict__ src, const float* __restrict__ vals,
    const int* __restrict__ sidx, const int* __restrict__ didx,
    float* __restrict__ dst, int nnz) {
  int e = (int)(blockIdx.x * (blockDim.x >> 5) + (threadIdx.x >> 5));
  if (e >= nnz) return;
  int lane = threadIdx.x & 31;
  int r = sidx[e];
  int c = didx[e];
  float v = vals[e];
  float4 x = ((const float4*)(src + (size_t)r * D))[lane];
  float* d = dst + (size_t)c * D + lane * 4;
  atomicAdd(d + 0, v * x.x);
  atomicAdd(d + 1, v * x.y);
  atomicAdd(d + 2, v * x.z);
  atomicAdd(d + 3, v * x.w);
}

// ------------------------------- LayerNorm(+residual) with WMMA row statistics
// One wave per 16-row tile. Row sums / sums-of-squares via
// V_WMMA_F32_16X16X4_F32 against an all-ones B (B layout irrelevant).
// A 16x4 f32 layout: lanes 0-15 hold K={k0,k0+1}, lanes 16-31 hold K={k0+2,k0+3}.
// C layout: VGPR r holds row r (lanes 0-15) and row r+8 (lanes 16-31).
__global__ __launch_bounds__(256) void ln_residual_wmma(
    float* __restrict__ h2, const float* __restrict__ ego,
    const float* __restrict__ gamma, const float* __restrict__ beta,
    int ntiles) {
  __shared__ float smu[8 * 16];
  __shared__ float srs[8 * 16];
  int wslot = threadIdx.x >> 5;
  int lane  = threadIdx.x & 31;
  int tile  = blockIdx.x * 8 + wslot;   // wave-uniform -> EXEC all-ones inside the branch

  if (tile < ntiles) {
    size_t base = (size_t)tile * 16 * D;
    const float* rowp = h2 + base + (size_t)(lane & 15) * D + ((lane >> 4) << 1);
    v8f csum = {};
    v8f csq  = {};
    v2f one2 = {1.0f, 1.0f};
#pragma unroll
    for (int s = 0; s < 32; ++s) {
      v2f a = *(const v2f*)(rowp + 4 * s);
      csum = __builtin_amdgcn_wmma_f32_16x16x4_f32(
          false, a, false, one2, (short)0, csum, false, false);
      v2f a2 = a * a;
      csq = __builtin_amdgcn_wmma_f32_16x16x4_f32(
          false, a2, false, one2, (short)0, csq, false, false);
    }
    // lane 0 holds stats for rows 0..7, lane 16 for rows 8..15 (all N equal)
    if ((lane & 15) == 0) {
      int rbase = (lane >> 4) * 8;
#pragma unroll
      for (int r = 0; r < 8; ++r) {
        float mu  = csum[r] * (1.0f / D);
        float var = csq[r] * (1.0f / D) - mu * mu;
        float rs  = rsqrtf(var + 1e-5f);
        smu[wslot * 16 + rbase + r] = mu;
        srs[wslot * 16 + rbase + r] = rs;
      }
    }
  }
  __syncthreads();

  if (tile < ntiles) {
    size_t base = (size_t)tile * 16 * D;
#pragma unroll
    for (int it = 0; it < 16; ++it) {
      int chunk = it * 32 + lane;         // 512 float4 chunks per tile
      int row   = chunk >> 5;             // 32 chunks per 128-float row
      int col   = (chunk & 31) << 2;
      size_t idx = base + (size_t)chunk * 4;
      float4 x = *(const float4*)(h2 + idx);
      float4 e = *(const float4*)(ego + idx);
      float4 g = *(const float4*)(gamma + col);
      float4 b = *(const float4*)(beta + col);
      float mu = smu[wslot * 16 + row];
      float rs = srs[wslot * 16 + row];
      float4 o;
      o.x = (x.x - mu) * rs * g.x + b.x + e.x;
      o.y = (x.y - mu) * rs * g.y + b.y + e.y;
      o.z = (x.z - mu) * rs * g.z + b.z + e.z;
      o.w = (x.w - mu) * rs * g.w + b.w + e.w;
      *(float4*)(h2 + idx) = o;           // in-place: element read by same lane above
    }
  }
}

// ---------------------------------------------------------------- launcher
extern "C" void kernel_launch(void* const* d_in, const int* in_sizes, int n_in,
                              void* d_out, int out_size, void* d_ws, size_t ws_size,
                              hipStream_t stream) {
  const float* ego   = (const float*)d_in[0];
  const float* vals  = (const float*)d_in[1];
  const float* gamma = (const float*)d_in[2];
  const float* beta  = (const float*)d_in[3];
  const int*   rows  = (const int*)d_in[4];
  const int*   cols  = (const int*)d_in[5];

  int nnz   = in_sizes[1];
  int nrows = in_sizes[0] / D;        // N = 150000 (divisible by 16)

  float* h  = (float*)d_ws;           // 76.8 MB scratch: first hop
  float* h2 = (float*)d_out;          // second hop accumulated directly in output

  long n4 = (long)nrows * (D / 4);
  int  zb = (int)((n4 + 255) / 256);
  zero_f4<<<zb, 256, 0, stream>>>((float4*)h, n4);
  zero_f4<<<zb, 256, 0, stream>>>((float4*)h2, n4);

  int sb = (nnz + 7) / 8;             // 8 waves (edges) per 256-thread block
  // pass 1: h[col] += v * ego[row]
  spmm_scatter<<<sb, 256, 0, stream>>>(ego, vals, rows, cols, h, nnz);
  // pass 2: h2[row] += v * h[col]
  spmm_scatter<<<sb, 256, 0, stream>>>(h, vals, cols, rows, h2, nnz);

  int ntiles = nrows / 16;            // 9375 tiles
  int lb = (ntiles + 7) / 8;
  ln_residual_wmma<<<lb, 256, 0, stream>>>(h2, ego, gamma, beta, ntiles);
}